// SelfAttention_54546084659541
// MI455X (gfx1250) — compile-verified
//
#include <hip/hip_runtime.h>
#include <hip/hip_bf16.h>
#include <math.h>

// ---------------------------------------------------------------------------
// FNO self-attention, reformulated entirely in truncated-mode space.
// All heavy stages are fp32 WMMA (V_WMMA_F32_16X16X4_F32) GEMMs on gfx1250.
// ---------------------------------------------------------------------------

typedef __attribute__((ext_vector_type(2))) float v2f;
typedef __attribute__((ext_vector_type(8))) float v8f;

__device__ __forceinline__ v8f wmma4(v2f a, v2f b, v8f c) {
  // D = A(16x4 f32) * B(4x16 f32) + C ; 8-arg pattern per CDNA5 probe doc
  return __builtin_amdgcn_wmma_f32_16x16x4_f32(false, a, false, b, (short)0, c,
                                               false, false);
}

#define PI2_64 0.0981747704246810387f // 2*pi/64

// cos/sin of 2*pi*((prod) mod 64)/64
__device__ __forceinline__ void tw(int prod, float* c, float* s) {
  float ang = (float)(prod & 63) * PI2_64;
  __sincosf(ang, s, c);
}

// ---------------------------------------------------------------------------
// K1: forward truncated rfft2 of v : [64 maps * 4ch] -> Vf[n][i][64 modes][2]
// ---------------------------------------------------------------------------
__global__ void k_fwd_modes_v(const float* __restrict__ v, float* __restrict__ Vf) {
  int map = blockIdx.x; // n*4 + i, 0..255
  const float* img = v + (size_t)map * 4096;
  __shared__ float simg[4096];
  __shared__ float T[16][64]; // rows: 2x=Re, 2x+1=Im
  for (int t = threadIdx.x; t < 4096; t += 256) simg[t] = img[t];
  __syncthreads();
  for (int e = threadIdx.x; e < 512; e += 256) {
    int x = e >> 6, c = e & 63;
    float tr = 0.f, ti = 0.f;
    for (int r = 0; r < 64; ++r) {
      float cs, sn; tw(x * r, &cs, &sn);         // e^{-i t}: (cs, -sn)
      float val = simg[r * 64 + c];
      tr += cs * val; ti -= sn * val;
    }
    T[2 * x][c] = tr; T[2 * x + 1][c] = ti;
  }
  __syncthreads();
  if (threadIdx.x < 64) {
    int x = threadIdx.x >> 3, y = threadIdx.x & 7;
    float mr = 0.f, mi = 0.f;
    for (int c = 0; c < 64; ++c) {
      float cs, sn; tw(y * c, &cs, &sn);         // multiply by e^{-i p}
      float trv = T[2 * x][c], tiv = T[2 * x + 1][c];
      mr += trv * cs + tiv * sn;
      mi += tiv * cs - trv * sn;
    }
    size_t o = ((size_t)map * 64 + threadIdx.x) * 2;
    Vf[o] = mr; Vf[o + 1] = mi;
  }
}

// ---------------------------------------------------------------------------
// K2: mode mixing -> Q/K/V tokens in [bh=16][tok=32][feat=4096] layout.
//     Ktok carries the Parseval weights (2 / 0.5 / 1,0) folded in.
// ---------------------------------------------------------------------------
__global__ void k_qkv_modes(const float* __restrict__ Vf,
                            const float* __restrict__ qwr, const float* __restrict__ qwi,
                            const float* __restrict__ kwr, const float* __restrict__ kwi,
                            const float* __restrict__ vwr, const float* __restrict__ vwi,
                            float* __restrict__ Qtok, float* __restrict__ Ktok,
                            float* __restrict__ Vtok) {
  int o = blockIdx.x;       // 0..255
  int n = blockIdx.y;       // 0..63
  int mode = threadIdx.x;   // 0..63
  int x = mode >> 3, y = mode & 7;
  float qr = 0, qi = 0, kr = 0, ki = 0, pr = 0, pi = 0;
  for (int i = 0; i < 4; ++i) {
    size_t vofs = (((size_t)(n * 4 + i)) * 64 + mode) * 2;
    float xr = Vf[vofs], xi = Vf[vofs + 1];
    size_t w = ((size_t)(i * 256 + o)) * 64 + mode;
    float wr, wi;
    wr = qwr[w]; wi = qwi[w]; qr += xr * wr - xi * wi; qi += xr * wi + xi * wr;
    wr = kwr[w]; wi = kwi[w]; kr += xr * wr - xi * wi; ki += xr * wi + xi * wr;
    wr = vwr[w]; wi = vwi[w]; pr += xr * wr - xi * wi; pi += xr * wi + xi * wr;
  }
  // reshape gymnastics: n=(b,s_orig); (s_orig*256+o) -> h*1024 + tok*32 + d
  int b = n >> 5, so = n & 31, m = so >> 2, j = so & 3;
  int g = o >> 5, d = o & 31;
  size_t base = (((size_t)(b * 8 + m) * 32 + (j * 8 + g)) * 4096) +
                (size_t)d * 128 + mode * 2;
  Qtok[base] = qr; Qtok[base + 1] = qi;
  float wR = 2.f, wI = 2.f;                 // Parseval weights
  if (y == 0) { if (x == 0) { wR = 1.f; wI = 0.f; } else { wR = 0.5f; wI = 0.5f; } }
  Ktok[base] = wR * kr; Ktok[base + 1] = wI * ki;
  Vtok[base] = pr; Vtok[base + 1] = pi;
}

// ---------------------------------------------------------------------------
// K3: Gram matrices via WMMA: l2[bh][32][32] = Q[32x4096] * Kw[32x4096]^T
// ---------------------------------------------------------------------------
__global__ void k_gram(const float* __restrict__ Qtok, const float* __restrict__ Ktok,
                       float* __restrict__ l2) {
  int bh = blockIdx.x;                 // 16
  int lane = threadIdx.x & 31;
  int wv = threadIdx.x >> 5;           // 4 waves -> 4 16x16 tiles
  int mt = wv >> 1, nt = wv & 1;
  int row = lane & 15;
  int khalf = (lane >> 4) * 2;
  const float* qp = Qtok + (size_t)bh * 131072 + (size_t)(mt * 16 + row) * 4096 + khalf;
  const float* kp = Ktok + (size_t)bh * 131072 + (size_t)(nt * 16 + row) * 4096 + khalf;
  v8f c = {};
#pragma unroll 4
  for (int kk = 0; kk < 4096; kk += 4) {
    v2f a = *(const v2f*)(qp + kk);
    v2f b = *(const v2f*)(kp + kk);
    c = wmma4(a, b, c);
  }
  const float fac = 0.5f / 16777216.0f; // scale * dx^2 * 1/(H*W)
  int N = nt * 16 + (lane & 15);
  int mb = (lane >> 4) * 8;
  for (int t = 0; t < 8; ++t)
    l2[((size_t)bh * 32 + mt * 16 + mb + t) * 32 + N] = c[t] * fac;
}

// ---------------------------------------------------------------------------
// K4: row softmax
// ---------------------------------------------------------------------------
__global__ void k_softmax(const float* __restrict__ l2, float* __restrict__ A) {
  int bh = blockIdx.x; int s = threadIdx.x; // 32 threads
  const float* row = l2 + ((size_t)bh * 32 + s) * 32;
  float mx = -1e30f;
  for (int t = 0; t < 32; ++t) mx = fmaxf(mx, row[t]);
  float e[32], sum = 0.f;
  for (int t = 0; t < 32; ++t) { e[t] = __expf(row[t] - mx); sum += e[t]; }
  float inv = 1.0f / sum;
  float* out = A + ((size_t)bh * 32 + s) * 32;
  for (int t = 0; t < 32; ++t) out[t] = e[t] * inv;
}

// ---------------------------------------------------------------------------
// K5: U = A x Vtok  (WMMA, K=32), scattered back to Un[n][ic=256][mode][2]
// ---------------------------------------------------------------------------
__global__ void k_applyA(const float* __restrict__ A, const float* __restrict__ Vtok,
                         float* __restrict__ Un) {
  int bx = blockIdx.x;  // 32
  int bh = blockIdx.y;  // 16
  int lane = threadIdx.x & 31, wv = threadIdx.x >> 5;
  __shared__ float sA[1024];
  for (int t = threadIdx.x; t < 1024; t += 128) sA[t] = A[(size_t)bh * 1024 + t];
  __syncthreads();
  int b = bh >> 3, m = bh & 7;
  const float* V = Vtok + (size_t)bh * 131072;
  int khalf = (lane >> 4) * 2;
  for (int i = 0; i < 4; ++i) {
    int tile = bx * 16 + wv * 4 + i;   // 512 tiles: 2 M x 256 N
    int mt = tile & 1, nbase = (tile >> 1) * 16;
    int arow = mt * 16 + (lane & 15);
    int bcol = nbase + (lane & 15);
    v8f c = {};
#pragma unroll
    for (int kk = 0; kk < 32; kk += 4) {
      v2f a, bb;
      a.x = sA[arow * 32 + kk + khalf];
      a.y = sA[arow * 32 + kk + khalf + 1];
      bb.x = V[(size_t)(kk + khalf) * 4096 + bcol];
      bb.y = V[(size_t)(kk + khalf + 1) * 4096 + bcol];
      c = wmma4(a, bb, c);
    }
    int d = bcol >> 7, rem = bcol & 127;
    for (int t = 0; t < 8; ++t) {
      int tok = mt * 16 + t + ((lane >> 4) ? 8 : 0);
      int n = b * 32 + m * 4 + (tok >> 3);
      int o = (tok & 7) * 32 + d;
      Un[((size_t)n * 256 + o) * 128 + rem] = c[t];
    }
  }
}

// ---------------------------------------------------------------------------
// K6: C1 = modes einsum(Un', u1w)  (Un' = rfft2(irfft2(Un)) halving rule),
//     S1 = s1w @ Un (1x1 conv folded through irfft2 linearity)
// ---------------------------------------------------------------------------
__global__ void k_c1s1(const float* __restrict__ Un, const float* __restrict__ u1wr,
                       const float* __restrict__ u1wi, const float* __restrict__ s1w,
                       float* __restrict__ C1, float* __restrict__ S1) {
  int n = blockIdx.x; // 64
  for (int e = threadIdx.x; e < 2048; e += 256) {
    int oc = e >> 6, mode = e & 63;
    int x = mode >> 3, y = mode & 7;
    float hr = 1.f, hi = 1.f;
    if (y == 0) { if (x == 0) hi = 0.f; else { hr = 0.5f; hi = 0.5f; } }
    float cr = 0, ci_ = 0, sr = 0, si = 0;
    const float* up = Un + (size_t)n * 32768 + mode * 2;
    const float* wrp = u1wr + (size_t)oc * 64 + mode;
    const float* wip = u1wi + (size_t)oc * 64 + mode;
    const float* swp = s1w + oc * 256;
    for (int ic = 0; ic < 256; ++ic) {
      float ur = up[(size_t)ic * 128], ui = up[(size_t)ic * 128 + 1];
      float wr = wrp[(size_t)ic * 2048], wi = wip[(size_t)ic * 2048];
      float ar = ur * hr, ai = ui * hi;
      cr += ar * wr - ai * wi; ci_ += ar * wi + ai * wr;
      float sw = swp[ic];
      sr += sw * ur; si += sw * ui;
    }
    size_t o = ((size_t)(n * 32 + oc) * 64 + mode) * 2;
    C1[o] = cr; C1[o + 1] = ci_;
    S1[o] = sr; S1[o + 1] = si;
  }
}

// ---------------------------------------------------------------------------
// Shared synthesis helpers: irfft2 of an 8x8 mode block as two WMMA GEMMs:
//   Z[64x16] = A1(row basis) x B1(modes, 16x16);  out = Z x W2(col basis)
// ---------------------------------------------------------------------------
__device__ __forceinline__ void synthA_lane(int r, int k0, v2f* a) {
  float c0, s0; tw((k0 >> 1) * r, &c0, &s0);
  a->x = ((k0 & 1) ? s0 : c0) * (1.0f / 64.0f);
  float c1, s1; tw(((k0 + 1) >> 1) * r, &c1, &s1);
  a->y = (((k0 + 1) & 1) ? s1 : c1) * (1.0f / 64.0f);
}
__device__ __forceinline__ void synthB_lane(int col, int k0, v2f* b) {
  int y0 = k0 >> 1; float c0, s0; tw(y0 * col, &c0, &s0);
  float cy0 = (y0 == 0) ? 1.0f : 2.0f;
  b->x = ((k0 & 1) ? -s0 : c0) * cy0 * (1.0f / 64.0f);
  int y1 = (k0 + 1) >> 1; float c1, s1; tw(y1 * col, &c1, &s1);
  float cy1 = (y1 == 0) ? 1.0f : 2.0f;
  b->y = (((k0 + 1) & 1) ? -s1 : c1) * cy1 * (1.0f / 64.0f);
}

// ---------------------------------------------------------------------------
// K7: x1 = gelu(irfft2(C1)) + irfft2(S1) + s1b   (WMMA synthesis, 2048 maps)
// ---------------------------------------------------------------------------
__global__ void k_synth_x1(const float* __restrict__ C1, const float* __restrict__ S1,
                           const float* __restrict__ s1b, float* __restrict__ x1) {
  int map = blockIdx.x;            // n*32 + ch
  int ch = map & 31;
  int lane = threadIdx.x & 31, wv = threadIdx.x >> 5;
  __shared__ float B1c[16][16], B1s[16][16];
  __shared__ float Zc[64][16], Zs[64][16];
  for (int e = threadIdx.x; e < 64; e += 128) {
    int x = e >> 3, y = e & 7;
    size_t o = ((size_t)map * 64 + e) * 2;
    float cr = C1[o], ci_ = C1[o + 1], sr = S1[o], si = S1[o + 1];
    B1c[2 * x][2 * y] = cr;   B1c[2 * x][2 * y + 1] = ci_;
    B1c[2 * x + 1][2 * y] = -ci_; B1c[2 * x + 1][2 * y + 1] = cr;
    B1s[2 * x][2 * y] = sr;   B1s[2 * x][2 * y + 1] = si;
    B1s[2 * x + 1][2 * y] = -si;  B1s[2 * x + 1][2 * y + 1] = sr;
  }
  __syncthreads();
  int khalf = (lane >> 4) * 2;
  { // GEMM-A: wave wv -> Z rows [16wv, 16wv+16)
    int r = wv * 16 + (lane & 15);
    v8f aC = {}, aS = {};
#pragma unroll
    for (int kk = 0; kk < 16; kk += 4) {
      int k0 = kk + khalf;
      v2f a; synthA_lane(r, k0, &a);
      v2f bc, bs;
      bc.x = B1c[k0][lane & 15]; bc.y = B1c[k0 + 1][lane & 15];
      bs.x = B1s[k0][lane & 15]; bs.y = B1s[k0 + 1][lane & 15];
      aC = wmma4(a, bc, aC);
      aS = wmma4(a, bs, aS);
    }
    int N = lane & 15, mb = (lane >> 4) * 8;
    for (int t = 0; t < 8; ++t) {
      Zc[wv * 16 + mb + t][N] = aC[t];
      Zs[wv * 16 + mb + t][N] = aS[t];
    }
  }
  __syncthreads();
  float bias = s1b[ch];
  float* outp = x1 + (size_t)map * 4096;
  for (int ct = 0; ct < 4; ++ct) { // GEMM-B: row tile wv, col tiles 0..3
    int col = ct * 16 + (lane & 15);
    int arow = wv * 16 + (lane & 15);
    v8f aC = {}, aS = {};
#pragma unroll
    for (int kk = 0; kk < 16; kk += 4) {
      int k0 = kk + khalf;
      v2f ac, as_, b;
      ac.x = Zc[arow][k0]; ac.y = Zc[arow][k0 + 1];
      as_.x = Zs[arow][k0]; as_.y = Zs[arow][k0 + 1];
      synthB_lane(col, k0, &b);
      aC = wmma4(ac, b, aC);
      aS = wmma4(as_, b, aS);
    }
    int N = ct * 16 + (lane & 15), mb = (lane >> 4) * 8;
    for (int t = 0; t < 8; ++t) {
      int r = wv * 16 + mb + t;
      float av = aC[t];
      float g = 0.5f * av * (1.0f + erff(av * 0.70710678118f)); // exact gelu
      outp[r * 64 + N] = g + aS[t] + bias;
    }
  }
}

// ---------------------------------------------------------------------------
// K8: truncated rfft2 of x1 via WMMA: T[16x64] = A2 x img, then y-contraction
// ---------------------------------------------------------------------------
__global__ void k_fwd_modes_x1(const float* __restrict__ x1, float* __restrict__ X1f) {
  int map = blockIdx.x; // 2048
  int lane = threadIdx.x & 31, wv = threadIdx.x >> 5;
  __shared__ float simg[4096];
  __shared__ float T[16][64];
  const float* img = x1 + (size_t)map * 4096;
  for (int t = threadIdx.x; t < 4096; t += 128) simg[t] = img[t];
  __syncthreads();
  {
    int M = lane & 15; int x = M >> 1;
    int khalf = (lane >> 4) * 2;
    int col = wv * 16 + (lane & 15);
    v8f acc = {};
#pragma unroll 4
    for (int kk = 0; kk < 64; kk += 4) {
      int k0 = kk + khalf;
      v2f a, b;
      { float c0, s0; tw(x * k0, &c0, &s0);       a.x = (M & 1) ? -s0 : c0; }
      { float c1, s1; tw(x * (k0 + 1), &c1, &s1); a.y = (M & 1) ? -s1 : c1; }
      b.x = simg[k0 * 64 + col];
      b.y = simg[(k0 + 1) * 64 + col];
      acc = wmma4(a, b, acc);
    }
    int mb = (lane >> 4) * 8;
    for (int t = 0; t < 8; ++t) T[mb + t][col] = acc[t];
  }
  __syncthreads();
  if (threadIdx.x < 64) {
    int x = threadIdx.x >> 3, y = threadIdx.x & 7;
    float mr = 0.f, mi = 0.f;
    for (int c = 0; c < 64; ++c) {
      float cs, sn; tw(y * c, &cs, &sn);
      float trv = T[2 * x][c], tiv = T[2 * x + 1][c];
      mr += trv * cs + tiv * sn;
      mi += tiv * cs - trv * sn;
    }
    size_t o = ((size_t)map * 64 + threadIdx.x) * 2;
    X1f[o] = mr; X1f[o + 1] = mi;
  }
}

// ---------------------------------------------------------------------------
// K9: C2 = einsum(X1f, u2w) in mode space (tiny)
// ---------------------------------------------------------------------------
__global__ void k_c2(const float* __restrict__ X1f, const float* __restrict__ u2wr,
                     const float* __restrict__ u2wi, float* __restrict__ C2) {
  int n = blockIdx.x;
  int oc = threadIdx.x >> 6, mode = threadIdx.x & 63;
  float cr = 0, ci_ = 0;
  for (int ic = 0; ic < 32; ++ic) {
    size_t xo = (((size_t)n * 32 + ic) * 64 + mode) * 2;
    float xr = X1f[xo], xi = X1f[xo + 1];
    size_t w = ((size_t)(ic * 4 + oc)) * 64 + mode;
    float wr = u2wr[w], wi = u2wi[w];
    cr += xr * wr - xi * wi; ci_ += xr * wi + xi * wr;
  }
  size_t o = (((size_t)n * 4 + oc) * 64 + mode) * 2;
  C2[o] = cr; C2[o + 1] = ci_;
}

// ---------------------------------------------------------------------------
// K10: out = irfft2(C2) + s2w@x1 + s2b   (WMMA synthesis + pointwise skip)
// ---------------------------------------------------------------------------
__global__ void k_synth_out(const float* __restrict__ C2, const float* __restrict__ x1,
                            const float* __restrict__ s2w, const float* __restrict__ s2b,
                            float* __restrict__ out) {
  int blk = blockIdx.x; // n*4 + oc, 256
  int n = blk >> 2, oc = blk & 3;
  int lane = threadIdx.x & 31, wv = threadIdx.x >> 5;
  __shared__ float B1[16][16];
  __shared__ float Z[64][16];
  __shared__ float sw[32];
  if (threadIdx.x < 32) sw[threadIdx.x] = s2w[oc * 32 + threadIdx.x];
  for (int e = threadIdx.x; e < 64; e += 128) {
    int x = e >> 3, y = e & 7;
    size_t o = ((size_t)blk * 64 + e) * 2;
    float cr = C2[o], ci_ = C2[o + 1];
    B1[2 * x][2 * y] = cr;   B1[2 * x][2 * y + 1] = ci_;
    B1[2 * x + 1][2 * y] = -ci_; B1[2 * x + 1][2 * y + 1] = cr;
  }
  __syncthreads();
  int khalf = (lane >> 4) * 2;
  {
    int r = wv * 16 + (lane & 15);
    v8f acc = {};
#pragma unroll
    for (int kk = 0; kk < 16; kk += 4) {
      int k0 = kk + khalf;
      v2f a; synthA_lane(r, k0, &a);
      v2f b; b.x = B1[k0][lane & 15]; b.y = B1[k0 + 1][lane & 15];
      acc = wmma4(a, b, acc);
    }
    int N = lane & 15, mb = (lane >> 4) * 8;
    for (int t = 0; t < 8; ++t) Z[wv * 16 + mb + t][N] = acc[t];
  }
  __syncthreads();
  float bias = s2b[oc];
  const float* x1n = x1 + (size_t)n * 32 * 4096;
  for (int ct = 0; ct < 4; ++ct) {
    int col = ct * 16 + (lane & 15);
    int arow = wv * 16 + (lane & 15);
    v8f acc = {};
#pragma unroll
    for (int kk = 0; kk < 16; kk += 4) {
      int k0 = kk + khalf;
      v2f a, b;
      a.x = Z[arow][k0]; a.y = Z[arow][k0 + 1];
      synthB_lane(col, k0, &b);
      acc = wmma4(a, b, acc);
    }
    int N = ct * 16 + (lane & 15), mb = (lane >> 4) * 8;
    for (int t = 0; t < 8; ++t) {
      int r = wv * 16 + mb + t;
      float skip = bias;
      for (int ic = 0; ic < 32; ++ic)
        skip += sw[ic] * x1n[(size_t)ic * 4096 + r * 64 + N];
      out[(size_t)blk * 4096 + r * 64 + N] = acc[t] + skip;
    }
  }
}

// ---------------------------------------------------------------------------
extern "C" void kernel_launch(void* const* d_in, const int* in_sizes, int n_in,
                              void* d_out, int out_size, void* d_ws, size_t ws_size,
                              hipStream_t stream) {
  (void)in_sizes; (void)n_in; (void)out_size; (void)ws_size;
  const float* v    = (const float*)d_in[0];
  const float* qwr  = (const float*)d_in[1];
  const float* qwi  = (const float*)d_in[2];
  const float* kwr  = (const float*)d_in[3];
  const float* kwi  = (const float*)d_in[4];
  const float* vwr  = (const float*)d_in[5];
  const float* vwi  = (const float*)d_in[6];
  const float* u1wr = (const float*)d_in[7];
  const float* u1wi = (const float*)d_in[8];
  const float* u2wr = (const float*)d_in[9];
  const float* u2wi = (const float*)d_in[10];
  const float* s1w  = (const float*)d_in[11];
  const float* s1b  = (const float*)d_in[12];
  const float* s2w  = (const float*)d_in[13];
  const float* s2b  = (const float*)d_in[14];
  float* out = (float*)d_out;
  float* ws = (float*)d_ws;
  // workspace layout (floats); total ~17.7M floats (~68 MB) <= ws_size expected
  size_t VF = 0;
  size_t QT = VF + 32768;
  size_t KT = QT + 2097152;
  size_t VT = KT + 2097152;
  size_t L2 = VT + 2097152;
  size_t AM = L2 + 16384;
  size_t UN = AM + 16384;
  size_t C1 = UN + 2097152;
  size_t S1 = C1 + 262144;
  size_t X1 = S1 + 262144;
  size_t XF = X1 + 8388608;
  size_t C2 = XF + 262144;

  k_fwd_modes_v<<<256, 256, 0, stream>>>(v, ws + VF);
  k_qkv_modes<<<dim3(256, 64), 64, 0, stream>>>(ws + VF, qwr, qwi, kwr, kwi,
                                                vwr, vwi, ws + QT, ws + KT, ws + VT);
  k_gram<<<16, 128, 0, stream>>>(ws + QT, ws + KT, ws + L2);
  k_softmax<<<16, 32, 0, stream>>>(ws + L2, ws + AM);
  k_applyA<<<dim3(32, 16), 128, 0, stream>>>(ws + AM, ws + VT, ws + UN);
  k_c1s1<<<64, 256, 0, stream>>>(ws + UN, u1wr, u1wi, s1w, ws + C1, ws + S1);
  k_synth_x1<<<2048, 128, 0, stream>>>(ws + C1, ws + S1, s1b, ws + X1);
  k_fwd_modes_x1<<<2048, 128, 0, stream>>>(ws + X1, ws + XF);
  k_c2<<<64, 256, 0, stream>>>(ws + XF, u2wr, u2wi, ws + C2);
  k_synth_out<<<256, 128, 0, stream>>>(ws + C2, ws + X1, s2w, s2b, out);
}